// EdgeDecoder_83245056131289
// MI455X (gfx1250) — compile-verified
//
#include <hip/hip_runtime.h>

typedef __attribute__((ext_vector_type(2))) float v2f;
typedef __attribute__((ext_vector_type(4))) float v4f;
typedef __attribute__((ext_vector_type(8))) float v8f;

#define H            64
#define ROWPAD       68                 // padded LDS row stride (floats): conflict-free b64 fragment reads
#define WAVES_PB     4
#define EDGES_PW     16
#define LDS_PW       (2 * 16 * ROWPAD)  // src rows + dst rows per wave (floats)

__global__ __launch_bounds__(128) void edge_decoder_wmma(
    const float* __restrict__ z_user,
    const float* __restrict__ z_movie,
    const float* __restrict__ rel_emb,      // [5, 64]
    const int*   __restrict__ eli,          // [2, E] (int32)
    float* __restrict__ out,                // [E]
    int E)
{
    __shared__ float lds[WAVES_PB * LDS_PW];    // 34,816 bytes

    const int lane = threadIdx.x & 31;
    const int wave = threadIdx.x >> 5;
    const int tile = blockIdx.x * WAVES_PB + wave;
    const int e0   = tile * EDGES_PW;
    const bool active = (e0 + EDGES_PW) <= E;   // E is a multiple of 16 here; guard anyway

    const int n    = lane & 15;                 // column / row-in-tile id
    const int koff = (lane >> 4) * 2;           // K sub-offset for upper half-wave
    const bool ncol = (n < 5);

    float* src_rows = &lds[wave * LDS_PW];
    float* dst_rows = src_rows + 16 * ROWPAD;

    // ---- Preload B fragments (rel_emb^T zero-padded to 16 cols), loop-invariant.
    // Unconditional b64 loads from a clamped row; zero columns n>=5 via select (no exec divergence).
    v2f bfr[16];
    {
        const float* bp = rel_emb + (ncol ? n : 0) * H + koff;
#pragma unroll
        for (int t = 0; t < 16; ++t) {
            v2f bv = *(const v2f*)(bp + 4 * t);
            bfr[t].x = ncol ? bv.x : 0.0f;
            bfr[t].y = ncol ? bv.y : 0.0f;
        }
    }

    // ---- Stage 16 user rows + 16 movie rows (64 f32 each) into LDS with b128 loads.
    // Indices loaded once per lane, broadcast per row via shfl (ds_bpermute).
    if (active) {
        const int ui0 = eli[e0 + n];            // lane n holds indices for edge n
        const int mi0 = eli[E + e0 + n];
#pragma unroll
        for (int j = 0; j < 8; ++j) {
            const int row = (j << 1) | (lane >> 4);   // 2 rows x 16 chunks per iter
            const int ui = __shfl(ui0, row);
            const int mi = __shfl(mi0, row);
            v4f a = *(const v4f*)(z_user  + (size_t)ui * H + n * 4);
            v4f b = *(const v4f*)(z_movie + (size_t)mi * H + n * 4);
            *(v4f*)(src_rows + row * ROWPAD + n * 4) = a;
            *(v4f*)(dst_rows + row * ROWPAD + n * 4) = b;
        }
    }
    __syncthreads();

    // ---- Build all 16 A fragments (z_src*z_dst k-slices) first: batch the ds_loads,
    // then run the WMMA chain back-to-back (only C-accumulate dependency).
    v2f aprod[16];
    {
        const float* s = src_rows + n * ROWPAD + koff;   // A row m == lane&15
        const float* d = dst_rows + n * ROWPAD + koff;
#pragma unroll
        for (int t = 0; t < 16; ++t) {
            v2f sv = *(const v2f*)(s + 4 * t);
            v2f dv = *(const v2f*)(d + 4 * t);
            aprod[t] = sv * dv;                          // v_pk_mul_f32
        }
    }

    v8f c = {};
#pragma unroll
    for (int t = 0; t < 16; ++t) {
        c = __builtin_amdgcn_wmma_f32_16x16x4_f32(false, aprod[t], false, bfr[t],
                                                  (short)0, c, false, false);
    }
    __syncthreads();   // WAR: about to reuse src_rows as the score slab

    // ---- Scatter scores (C layout: VGPR v -> M=v (+8 for upper lanes), lane -> N) to LDS.
    float* sc = src_rows;                   // 16 edges * 9 floats (stride 9: conflict-free)
    const int mbase = (lane >> 4) << 3;
#pragma unroll
    for (int v = 0; v < 8; ++v) {
        if (ncol) sc[(v + mbase) * 9 + n] = c[v];
    }
    __syncthreads();

    // ---- Softmax over 5 relations + expected rating, one edge per lane (lanes 0-15).
    if (active && lane < 16) {
        const float* s = sc + lane * 9;
        float s0 = s[0], s1 = s[1], s2 = s[2], s3 = s[3], s4 = s[4];
        float mx = fmaxf(fmaxf(fmaxf(s0, s1), fmaxf(s2, s3)), s4);
        float x0 = __expf(s0 - mx);
        float x1 = __expf(s1 - mx);
        float x2 = __expf(s2 - mx);
        float x3 = __expf(s3 - mx);
        float x4 = __expf(s4 - mx);
        float denom = x0 + x1 + x2 + x3 + x4;
        float num   = x1 + 2.0f * x2 + 3.0f * x3 + 4.0f * x4;
        out[e0 + lane] = num / denom;
    }
}

extern "C" void kernel_launch(void* const* d_in, const int* in_sizes, int n_in,
                              void* d_out, int out_size, void* d_ws, size_t ws_size,
                              hipStream_t stream) {
    const float* z_user  = (const float*)d_in[0];
    const float* z_movie = (const float*)d_in[1];
    const float* rel_emb = (const float*)d_in[2];
    const int*   eli     = (const int*)d_in[3];
    float*       out     = (float*)d_out;

    const int E      = in_sizes[3] / 2;              // [2, E]
    const int tiles  = (E + EDGES_PW - 1) / EDGES_PW;
    const int blocks = (tiles + WAVES_PB - 1) / WAVES_PB;

    edge_decoder_wmma<<<blocks, WAVES_PB * 32, 0, stream>>>(
        z_user, z_movie, rel_emb, eli, out, E);
}